// HeteroGNNBlock_4217657885251
// MI455X (gfx1250) — compile-verified
//
#include <hip/hip_runtime.h>
#include <hip/hip_bf16.h>

typedef __attribute__((ext_vector_type(2))) float v2f;
typedef __attribute__((ext_vector_type(8))) float v8f;

#define D128 128
#define LDW 132   // padded LDS row stride (floats) -> conflict-free b64 reads

// ---------------------------------------------------------------------------
// zero workspace
// ---------------------------------------------------------------------------
__global__ void zero_kernel(float* __restrict__ p, long n) {
    long i = (long)blockIdx.x * blockDim.x + threadIdx.x;
    long stride = (long)gridDim.x * blockDim.x;
    for (; i < n; i += stride) p[i] = 0.0f;
}

// ---------------------------------------------------------------------------
// edge scatter: one wave per edge. agg[dst] += x_src[src]; cnt[dst] += 1
// ---------------------------------------------------------------------------
__global__ void scatter_kernel(const float* __restrict__ xsrc,
                               const int* __restrict__ ei, int E,
                               float* __restrict__ agg, float* __restrict__ cnt) {
    int g = blockIdx.x * blockDim.x + threadIdx.x;
    int e = g >> 5;
    int lane = g & 31;
    if (e >= E) return;
    int src = ei[e];        // row 0 of [2,E]
    int dst = ei[E + e];    // row 1 of [2,E]
    const float4* sp = (const float4*)(xsrc + (size_t)src * D128);
    float4 v = sp[lane];                      // 32 lanes x 16B = full 512B row
    float* dp = agg + (size_t)dst * D128 + lane * 4;
    atomicAdd(dp + 0, v.x);
    atomicAdd(dp + 1, v.y);
    atomicAdd(dp + 2, v.z);
    atomicAdd(dp + 3, v.w);
    if (lane == 0) atomicAdd(cnt + dst, 1.0f);
}

// ---------------------------------------------------------------------------
// SAGE GEMM: out[16-row tile] = (agg/max(cnt,1)) @ Wl^T + bl + xdst @ Wr^T
// One wave per 16-row tile, 8 waves/block, Wl/Wr staged in LDS.
// Uses V_WMMA_F32_16X16X4_F32 (fp32-exact), 32 K-steps x 8 N-tiles x 2 GEMMs.
// ---------------------------------------------------------------------------
template <int ACC>
__global__ void sage_gemm_kernel(const float* __restrict__ agg,
                                 const float* __restrict__ cnt,
                                 const float* __restrict__ xdst,
                                 const float* __restrict__ Wl,
                                 const float* __restrict__ bl,
                                 const float* __restrict__ Wr,
                                 float* __restrict__ out, int nrows) {
    __shared__ float sW[2 * D128 * LDW];   // 135168 B (< 320KB WGP LDS)
    float* sWl = sW;
    float* sWr = sW + D128 * LDW;

    for (int i = threadIdx.x; i < D128 * D128; i += blockDim.x) {
        int r = i >> 7, c = i & 127;
        sWl[r * LDW + c] = Wl[i];
        sWr[r * LDW + c] = Wr[i];
    }
    __syncthreads();

    const int lane = threadIdx.x & 31;
    const int wave = threadIdx.x >> 5;
    const int tile = blockIdx.x * 8 + wave;        // 16-row tile id (wave-uniform)
    if (tile * 16 >= nrows) return;                // uniform exit: EXEC stays all-1s

    const int m0   = tile * 16;
    const int idx  = lane & 15;
    const int half = lane >> 4;
    const int koff = 2 * half;                     // A/B frag K sub-offset per lane half
    const int row  = m0 + idx;

    v8f acc[8] = {};                               // 8 N-tiles of 16 cols each

    // ---- part 1: mean aggregation @ Wl^T ----
    const float rc = 1.0f / fmaxf(cnt[row], 1.0f);
    const float* arow = agg + (size_t)row * D128;
    for (int kt = 0; kt < 32; ++kt) {
        const int kk = kt * 4 + koff;
        v2f a;
        a.x = arow[kk] * rc;
        a.y = arow[kk + 1] * rc;
#pragma unroll
        for (int nt = 0; nt < 8; ++nt) {
            const float* wp = sWl + (size_t)(nt * 16 + idx) * LDW + kk;
            v2f b; b.x = wp[0]; b.y = wp[1];
            acc[nt] = __builtin_amdgcn_wmma_f32_16x16x4_f32(
                false, a, false, b, (short)0, acc[nt], false, false);
        }
    }

    // ---- part 2: x_dst @ Wr^T ----
    const float* xrow = xdst + (size_t)row * D128;
    for (int kt = 0; kt < 32; ++kt) {
        const int kk = kt * 4 + koff;
        v2f a;
        a.x = xrow[kk];
        a.y = xrow[kk + 1];
#pragma unroll
        for (int nt = 0; nt < 8; ++nt) {
            const float* wp = sWr + (size_t)(nt * 16 + idx) * LDW + kk;
            v2f b; b.x = wp[0]; b.y = wp[1];
            acc[nt] = __builtin_amdgcn_wmma_f32_16x16x4_f32(
                false, a, false, b, (short)0, acc[nt], false, false);
        }
    }

    // ---- epilogue: + bias, optional accumulate, store ----
    // C layout: vreg r -> global row m0 + r + 8*half, col n0 + idx
#pragma unroll
    for (int nt = 0; nt < 8; ++nt) {
        const int n = nt * 16 + idx;
        const float bias = bl[n];
        float* obase = out + (size_t)(m0 + 8 * half) * D128 + n;
#pragma unroll
        for (int r = 0; r < 8; ++r) {
            float v = acc[nt][r] + bias;
            if (ACC) v += obase[(size_t)r * D128];
            obase[(size_t)r * D128] = v;
        }
    }
}

// ---------------------------------------------------------------------------
// LayerNorm + residual + ReLU, in-place on d_out. One wave per row.
// rows [0,N): type A (gA/betaA, residual xA); rows [N,2N): type B.
// ---------------------------------------------------------------------------
__global__ void ln_res_relu_kernel(float* __restrict__ out,
                                   const float* __restrict__ xA,
                                   const float* __restrict__ xB,
                                   const float* __restrict__ gA,
                                   const float* __restrict__ bA,
                                   const float* __restrict__ gB,
                                   const float* __restrict__ bB, int N) {
    int g = blockIdx.x * blockDim.x + threadIdx.x;
    int row = g >> 5;
    int lane = g & 31;
    if (row >= 2 * N) return;

    const float* res;
    const float* gamma;
    const float* beta;
    if (row < N) { res = xA + (size_t)row * D128;       gamma = gA; beta = bA; }
    else         { res = xB + (size_t)(row - N) * D128; gamma = gB; beta = bB; }

    float* o = out + (size_t)row * D128;
    float4 v = ((const float4*)o)[lane];

    float s = v.x + v.y + v.z + v.w;
    float q = v.x * v.x + v.y * v.y + v.z * v.z + v.w * v.w;
#pragma unroll
    for (int off = 16; off > 0; off >>= 1) {
        s += __shfl_xor(s, off);
        q += __shfl_xor(q, off);
    }
    const float mu   = s * (1.0f / 128.0f);
    const float var  = q * (1.0f / 128.0f) - mu * mu;
    const float rstd = rsqrtf(var + 1e-5f);

    float4 gm = ((const float4*)gamma)[lane];
    float4 bt = ((const float4*)beta)[lane];
    float4 rx = ((const float4*)res)[lane];

    float4 y;
    y.x = fmaxf((v.x - mu) * rstd * gm.x + bt.x + rx.x, 0.0f);
    y.y = fmaxf((v.y - mu) * rstd * gm.y + bt.y + rx.y, 0.0f);
    y.z = fmaxf((v.z - mu) * rstd * gm.z + bt.z + rx.z, 0.0f);
    y.w = fmaxf((v.w - mu) * rstd * gm.w + bt.w + rx.w, 0.0f);
    ((float4*)o)[lane] = y;
}

// ---------------------------------------------------------------------------
extern "C" void kernel_launch(void* const* d_in, const int* in_sizes, int n_in,
                              void* d_out, int out_size, void* d_ws, size_t ws_size,
                              hipStream_t stream) {
    const float* xA    = (const float*)d_in[0];
    const float* xB    = (const float*)d_in[1];
    const int*   ei_ab = (const int*)d_in[2];
    const int*   ei_ba = (const int*)d_in[3];
    const int*   ei_aa = (const int*)d_in[4];
    const float* Wl_ab = (const float*)d_in[5];
    const float* bl_ab = (const float*)d_in[6];
    const float* Wr_ab = (const float*)d_in[7];
    const float* Wl_ba = (const float*)d_in[8];
    const float* bl_ba = (const float*)d_in[9];
    const float* Wr_ba = (const float*)d_in[10];
    const float* Wl_aa = (const float*)d_in[11];
    const float* bl_aa = (const float*)d_in[12];
    const float* Wr_aa = (const float*)d_in[13];
    const float* gA    = (const float*)d_in[14];
    const float* betaA = (const float*)d_in[15];
    const float* gB    = (const float*)d_in[16];
    const float* betaB = (const float*)d_in[17];

    const int N = in_sizes[0] / D128;
    const int E = in_sizes[2] / 2;

    float* outA = (float*)d_out;
    float* outB = (float*)d_out + (size_t)N * D128;

    float* agg = (float*)d_ws;                       // [N,128]
    float* cnt = agg + (size_t)N * D128;             // [N]
    const long nzero = (long)N * D128 + N;

    const int tiles     = (N + 15) / 16;
    const int gemm_grid = (tiles + 7) / 8;
    const int scat_grid = (E + 7) / 8;
    const int ln_grid   = (2 * N + 7) / 8;

    // ---- pass 1: B -> A (init outA) ----
    zero_kernel<<<2048, 256, 0, stream>>>(agg, nzero);
    scatter_kernel<<<scat_grid, 256, 0, stream>>>(xB, ei_ba, E, agg, cnt);
    sage_gemm_kernel<0><<<gemm_grid, 256, 0, stream>>>(agg, cnt, xA, Wl_ba, bl_ba, Wr_ba, outA, N);

    // ---- pass 2: A -> A (accumulate into outA) ----
    zero_kernel<<<2048, 256, 0, stream>>>(agg, nzero);
    scatter_kernel<<<scat_grid, 256, 0, stream>>>(xA, ei_aa, E, agg, cnt);
    sage_gemm_kernel<1><<<gemm_grid, 256, 0, stream>>>(agg, cnt, xA, Wl_aa, bl_aa, Wr_aa, outA, N);

    // ---- pass 3: A -> B (init outB) ----
    zero_kernel<<<2048, 256, 0, stream>>>(agg, nzero);
    scatter_kernel<<<scat_grid, 256, 0, stream>>>(xA, ei_ab, E, agg, cnt);
    sage_gemm_kernel<0><<<gemm_grid, 256, 0, stream>>>(agg, cnt, xB, Wl_ab, bl_ab, Wr_ab, outB, N);

    // ---- pass 4: LayerNorm + residual + ReLU ----
    ln_res_relu_kernel<<<ln_grid, 256, 0, stream>>>((float*)d_out, xA, xB, gA, betaA, gB, betaB, N);
}